// CTCLoss_70935679861266
// MI455X (gfx1250) — compile-verified
//
#include <hip/hip_runtime.h>
#include <math.h>

// Problem constants (fixed by the reference setup)
#define T_STEPS 1024
#define B_SZ    128
#define C_SZ    256
#define S_SZ    64
#define L_EXT   (2 * S_SZ + 1)   // 129 lattice states
#define CH      8                // time rows per TDM chunk
#define NCHUNK  (T_STEPS / CH)   // 128 chunks
#define NEGV    (-1e30f)
#define K_PER   5                // states per lane: s = lane + 32k

typedef __attribute__((ext_vector_type(4))) unsigned int u32x4;
typedef __attribute__((ext_vector_type(8))) int          i32x8;
typedef __attribute__((ext_vector_type(4))) int          i32x4;

#if __has_builtin(__builtin_amdgcn_tensor_load_to_lds) && __has_builtin(__builtin_amdgcn_s_wait_tensorcnt)
#define USE_TDM 1
#else
#define USE_TDM 0
#endif

#if USE_TDM
#define WAIT_TENSOR(n) __builtin_amdgcn_s_wait_tensorcnt(n)
#else
#define WAIT_TENSOR(n) ((void)0)
#endif

// 3-way logaddexp; feeding NEGV for c degenerates exactly to the 2-way form
// (exp underflows to 0), so one transcendental chain serves both skip cases.
__device__ __forceinline__ float lae3(float a, float b, float c) {
    float m = fmaxf(fmaxf(a, b), c);
    return m + __logf(__expf(a - m) + __expf(b - m) + __expf(c - m));
}
__device__ __forceinline__ float lae2(float a, float b) {
    float m = fmaxf(a, b);
    return m + __logf(__expf(a - m) + __expf(b - m));
}

// One wave32 per batch element. Alpha lives entirely in registers (5 states
// per lane); neighbor states alpha[s-1], alpha[s-2] come from wave shuffles.
// TDM streams log_probs rows (2-D strided tiles: CH rows x C_SZ floats, row
// stride B*C) into a double-buffered LDS ring overlapped via s_wait_tensorcnt.
__global__ __launch_bounds__(32) void ctc_alpha_kernel(
    const float* __restrict__ log_probs,      // (T, B, C)
    const int*   __restrict__ targets,        // (B, S)
    const int*   __restrict__ target_lengths, // (B,)
    float*       __restrict__ per_b)          // (B,) loss/len per element
{
    const int b    = blockIdx.x;
    const int lane = threadIdx.x;

    __shared__ float rows[2][CH][C_SZ];      // 16 KB TDM ring (only LDS left)

    // ---- per-lane extended-label classes and skip flags -------------------
    int  ext[K_PER];
    bool skp[K_PER];
#pragma unroll
    for (int k = 0; k < K_PER; ++k) {
        const int s = lane + 32 * k;
        int e = 0;
        bool sk = false;
        if (s < L_EXT && (s & 1)) {
            const int j = s >> 1;
            e = targets[b * S_SZ + j];
            if (s >= 3) sk = (e != targets[b * S_SZ + j - 1]);
        }
        ext[k] = e;
        skp[k] = sk;
    }

    const unsigned rows_base = (unsigned)(uintptr_t)&rows[0][0][0];

    // ---- chunk loader: TDM 2-D strided tile (CH rows x C_SZ floats) -------
    auto loadChunk = [&](int slot, int c) {
#if USE_TDM
        unsigned long long ga = (unsigned long long)(uintptr_t)log_probs +
            (((unsigned long long)(c * CH) * B_SZ + (unsigned long long)b) *
             (unsigned long long)C_SZ) * 4ull;
        u32x4 g0;
        g0.x = 1u;                                          // count = 1 valid D#
        g0.y = rows_base + (unsigned)slot * CH * C_SZ * 4u; // lds_addr (bytes)
        g0.z = (unsigned)(ga & 0xFFFFFFFFu);                // global_addr[31:0]
        g0.w = (unsigned)((ga >> 32) & 0x01FFFFFFu)         // global_addr[56:32]
             | (2u << 30);                                  // type = 2 (image)
        i32x8 g1;
        g1[0] = (int)(2u << 16);             // data_size = 4 bytes
        g1[1] = (int)((unsigned)C_SZ << 16); // tensor_dim0[15:0] (bits 63:48)
        g1[2] = (int)((unsigned)CH   << 16); // tensor_dim1[15:0] (bits 95:80)
        g1[3] = (int)((unsigned)C_SZ << 16); // tile_dim0 (bits 127:112)
        g1[4] = (int)CH;                     // tile_dim1
        g1[5] = (int)(B_SZ * C_SZ);          // tensor_dim0_stride (elements)
        g1[6] = 0;
        g1[7] = 0;
        i32x4 gz4 = {0, 0, 0, 0};            // groups 2/3 unused for 2-D tile
        i32x8 gz8 = {0, 0, 0, 0, 0, 0, 0, 0};
        __builtin_amdgcn_tensor_load_to_lds(g0, g1, gz4, gz4, gz8, 0);
#else
        // Fallback: synchronous wave-cooperative float4 copy.
        for (int r = 0; r < CH; ++r) {
            const float4* src = (const float4*)(log_probs +
                ((size_t)(c * CH + r) * B_SZ + (size_t)b) * C_SZ);
            float4* dst = (float4*)&rows[slot][r][0];
            for (int i = lane; i < C_SZ / 4; i += 32) dst[i] = src[i];
        }
#endif
    };

    loadChunk(0, 0);
    loadChunk(1, 1);
    WAIT_TENSOR(1);   // chunk 0 landed (TDM completes in-order per wave)

    // ---- t = 0 init: alpha0[0] = lp[blank], alpha0[1] = lp[ext[1]] --------
    float a[K_PER];
#pragma unroll
    for (int k = 0; k < K_PER; ++k) {
        const int s = lane + 32 * k;
        float v = NEGV;
        if (s == 0) v = rows[0][0][0];
        else if (s == 1) v = rows[0][0][ext[k]];
        a[k] = v;
    }

    // ---- one recursion step: neighbor alphas via wave32 shuffles ----------
    auto step = [&](const float* __restrict__ row) {
        float na[K_PER];
#pragma unroll
        for (int k = 0; k < K_PER; ++k) {
            float up1 = __shfl_up(a[k], 1);                 // lanes >= 1
            float up2 = __shfl_up(a[k], 2);                 // lanes >= 2
            float c31 = (k > 0) ? __shfl(a[k - 1], 31) : NEGV;
            float c30 = (k > 0) ? __shfl(a[k - 1], 30) : NEGV;
            float am1 = (lane >= 1) ? up1 : c31;            // alpha[s-1]
            float am2 = (lane >= 2) ? up2                   // alpha[s-2]
                                    : ((lane == 1) ? c31 : c30);
            am2 = skp[k] ? am2 : NEGV;                      // skip gate on input
            na[k] = lae3(a[k], am1, am2) + row[ext[k]];
        }
#pragma unroll
        for (int k = 0; k < K_PER; ++k) a[k] = na[k];
    };

    // ---- chunk 0, rows 1..7 ----------------------------------------------
#pragma unroll
    for (int r = 1; r < CH; ++r) step(&rows[0][r][0]);

    // ---- steady state: prefetch chunk c+1 while consuming chunk c ---------
    for (int c = 1; c < NCHUNK; ++c) {
        const int slot = c & 1;
        if (c + 1 < NCHUNK) {
            loadChunk(slot ^ 1, c + 1);  // that slot's old data fully consumed
            WAIT_TENSOR(1);              // oldest outstanding (chunk c) done
        } else {
            WAIT_TENSOR(0);
        }
        const float* base = &rows[slot][0][0];
#pragma unroll
        for (int r = 0; r < CH; ++r) step(base + r * C_SZ);
    }

    // ---- terminal states via uniform shuffle + zero_infinity --------------
    auto getA = [&](int sidx) -> float {   // all 32 lanes execute (uniform)
        const int kk = sidx >> 5;
        float v = a[0];
        if (kk == 1) v = a[1];
        else if (kk == 2) v = a[2];
        else if (kk == 3) v = a[3];
        else if (kk == 4) v = a[4];
        return __shfl(v, sidx & 31);
    };
    const int len = target_lengths[b];
    const int idx = 2 * len;               // in [2, 128]
    float loss = -lae2(getA(idx), getA(idx - 1));
    if (!(fabsf(loss) < 1e29f)) loss = 0.0f;   // NaN/Inf/saturated -> 0
    if (lane == 0) per_b[b] = loss / (float)len;
}

// Deterministic fixed-order reduction: mean over B of per-sample losses.
__global__ __launch_bounds__(B_SZ) void ctc_reduce_kernel(
    const float* __restrict__ per_b, float* __restrict__ out)
{
    __shared__ float sm[B_SZ];
    const int t = threadIdx.x;
    sm[t] = per_b[t];
    __syncthreads();
    for (int st = B_SZ / 2; st > 0; st >>= 1) {
        if (t < st) sm[t] += sm[t + st];
        __syncthreads();
    }
    if (t == 0) out[0] = sm[0] / (float)B_SZ;
}

extern "C" void kernel_launch(void* const* d_in, const int* in_sizes, int n_in,
                              void* d_out, int out_size, void* d_ws, size_t ws_size,
                              hipStream_t stream) {
    (void)in_sizes; (void)n_in; (void)out_size; (void)ws_size;
    const float* log_probs      = (const float*)d_in[0];
    const int*   targets        = (const int*)d_in[1];
    const int*   target_lengths = (const int*)d_in[2];
    float* per_b = (float*)d_ws;          // B floats of scratch
    float* out   = (float*)d_out;

    ctc_alpha_kernel<<<B_SZ, 32, 0, stream>>>(log_probs, targets,
                                              target_lengths, per_b);
    ctc_reduce_kernel<<<1, B_SZ, 0, stream>>>(per_b, out);
}